// BandPass_28552942583884
// MI455X (gfx1250) — compile-verified
//
#include <hip/hip_runtime.h>

typedef __attribute__((ext_vector_type(2))) float v2f;
typedef __attribute__((ext_vector_type(4))) float v4f;
typedef __attribute__((ext_vector_type(8))) float v8f;

constexpr int Bt      = 128;
constexpr int Tt      = 200000;
constexpr int SEG_LEN = 800;             // payload samples per segment
constexpr int NSEG    = Tt / SEG_LEN;    // 250 segments (exact)
constexpr int WARM    = 160;             // warm-up samples (pole^160 ~ 1e-13)
constexpr int NTILE   = Bt / 16;         // 8 batch tiles of 16
constexpr int NWARMB  = WARM / 16;       // 10 warm-up blocks
constexpr int NPAYB   = SEG_LEN / 16;    // 50 payload blocks
constexpr int PFDIST  = 256;             // prefetch distance in samples (16 blocks)

// One wave32 per (batch-tile, time-segment).
// Per 16-sample block:  Y(16x16) = G(16x20) * [X(18x16); y_state(2x16)]
// via 5x V_WMMA_F32_16X16X4_F32 chained on the f32 accumulator.
__global__ __launch_bounds__(32) void bandpass_biquad_wmma(
    const float* __restrict__ x, float* __restrict__ out)
{
    __shared__ float hv[18];             // AR impulse response h[0..17]

    const int lane = threadIdx.x;        // wave32
    const int m    = lane & 15;          // A-matrix row (time) / batch column
    const int kh   = lane >> 4;          // which K-half of each K=4 chunk
    const int tile = blockIdx.x % NTILE;
    const int seg  = blockIdx.x / NTILE;

    // ---- normalized biquad coefficients (bandpass, const_skirt_gain=false) ----
    const double w0    = 2.0 * 3.14159265358979323846 * 2000.0 / 48000.0;
    const double alpha = sin(w0) / (2.0 * 0.707);
    const double a0d   = 1.0 + alpha;
    const float nb0 = (float)( alpha / a0d);
    const float nb1 = 0.0f;
    const float nb2 = (float)(-alpha / a0d);
    const float na1 = (float)(-2.0 * cos(w0) / a0d);
    const float na2 = (float)((1.0 - alpha) / a0d);

    if (lane == 0) {
        float hm2 = 0.f, hm1 = 0.f;
        for (int k = 0; k < 18; ++k) {
            float hk = (k == 0 ? 1.f : 0.f) - na1 * hm1 - na2 * hm2;
            hv[k] = hk; hm2 = hm1; hm1 = hk;
        }
    }
    __syncthreads();

    auto hat = [&](int k) -> float { return (k >= 0 && k < 18) ? hv[k] : 0.f; };

    // ---- build A operand: G[16 x 20] in 5 chunks of K=4 ----
    // ISA f32 A layout: vgpr0 = {K=4c+0 (lanes0-15), K=4c+2 (lanes16-31)},
    //                   vgpr1 = {K=4c+1, K=4c+3}; rows M = lane&15.
    v2f A[5];
#pragma unroll
    for (int c = 0; c < 5; ++c) {
        float e[2];
#pragma unroll
        for (int t = 0; t < 2; ++t) {
            const int j = 4 * c + 2 * kh + t;   // column of G
            float v;
            if (j == 18)      v = hat(m + 1);           // p[m]: weight of y[-1]
            else if (j == 19) v = -na2 * hat(m);        // q[m]: weight of y[-2]
            else {                                      // weight of x[t0-2+j]
                const int i = j - 2;
                v = nb2 * hat(m - i - 2);
                if (i >= -1) v += nb1 * hat(m - i - 1);
                if (i >= 0)  v += nb0 * hat(m - i);
            }
            e[t] = v;
        }
        A[c] = (v2f){e[0], e[1]};
    }

    // ---- per-wave row pointers / time range ----
    const float* __restrict__ xrow = x   + (size_t)(tile * 16 + m) * Tt;
    float*       __restrict__ yrow = out + (size_t)(tile * 16 + m) * Tt;
    const int outStart = seg * SEG_LEN;
    const int nwarm    = (seg == 0) ? 0 : NWARMB;
    int t = outStart - 16 * nwarm;       // first block start

    // B layout mirrors A: vgpr0 = rows {4c+0 | 4c+2}, vgpr1 = rows {4c+1 | 4c+3},
    // columns N = lane&15.  Row j of B = x[t-2+j]; rows 18/19 = y[-1]/y[-2].
    // NOTE: xb[4] is loaded by ALL lanes (lane n and n+16 hit the same address,
    // so the pair coalesces to one 8B request); kh==1 lanes discard it via the
    // state cndmask in compute().  This keeps the loop free of exec-mask edits.
    auto loadBlk = [&](int tb, v2f xb[5]) {       // unguarded (tb >= 2)
        const int base = tb - 2 + 2 * kh;
#pragma unroll
        for (int c = 0; c < 4; ++c) xb[c] = *(const v2f*)(xrow + base + 4 * c);
        xb[4] = *(const v2f*)(xrow + tb + 14);
    };

    float y15 = 0.f, y14 = 0.f;   // state (valid on lanes 16-31 = B rows 18/19)
    v2f xb[5], nx[5];
    v8f acc;

    auto compute = [&](void) {
        const v2f b4 = kh ? (v2f){y15, y14} : xb[4];
        v8f a = {0.f,0.f,0.f,0.f,0.f,0.f,0.f,0.f};
        a = __builtin_amdgcn_wmma_f32_16x16x4_f32(false, A[0], false, xb[0], (short)0, a, false, false);
        a = __builtin_amdgcn_wmma_f32_16x16x4_f32(false, A[1], false, xb[1], (short)0, a, false, false);
        a = __builtin_amdgcn_wmma_f32_16x16x4_f32(false, A[2], false, xb[2], (short)0, a, false, false);
        a = __builtin_amdgcn_wmma_f32_16x16x4_f32(false, A[3], false, xb[3], (short)0, a, false, false);
        a = __builtin_amdgcn_wmma_f32_16x16x4_f32(false, A[4], false, b4,    (short)0, a, false, false);
        acc = a;
        y15 = a[7];               // lanes 16-31: Y[15][n]
        y14 = a[6];               // lanes 16-31: Y[14][n]
    };
    auto store = [&](int tb) {
        // lane stores 8 contiguous samples of its batch row (aligned b128 x2)
        float* p = yrow + tb + 8 * kh;
        v4f lo = {acc[0], acc[1], acc[2], acc[3]};
        v4f hi = {acc[4], acc[5], acc[6], acc[7]};
        *(v4f*)p       = lo;
        *((v4f*)p + 1) = hi;
    };
    auto rotate = [&](void) {
#pragma unroll
        for (int c = 0; c < 5; ++c) xb[c] = nx[c];
    };

    // ---- initial block load (only place that can touch t-2 < 0) ----
    {
        const int base = t - 2 + 2 * kh;
#pragma unroll
        for (int c = 0; c < 4; ++c) {
            const int tt = base + 4 * c;
            xb[c] = (tt >= 0) ? *(const v2f*)(xrow + tt) : (v2f){0.f, 0.f};
        }
        xb[4] = *(const v2f*)(xrow + t + 14);
    }

    // ---- warm-up blocks: compute only, no store (uniform trip count) ----
    for (int blk = 0; blk < nwarm; ++blk, t += 16) {
        loadBlk(t + 16, nx);      // t+16 <= outStart < Tt
        compute();
        rotate();
    }

    // ---- payload blocks: unconditional load/compute/store; last block peeled ----
    for (int blk = 0; blk < NPAYB - 1; ++blk, t += 16) {
        loadBlk(t + 16, nx);
        {   // branch-free clamped prefetch of the x stream (speculative-safe)
            int tpf = t + PFDIST;
            tpf = (tpf > Tt - 16) ? (Tt - 16) : tpf;
            __builtin_prefetch(xrow + tpf, 0, 1);   // global_prefetch_b8
        }
        compute();
        store(t);
        rotate();
    }
    // final block: no next load
    compute();
    store(t);
}

extern "C" void kernel_launch(void* const* d_in, const int* in_sizes, int n_in,
                              void* d_out, int out_size, void* d_ws, size_t ws_size,
                              hipStream_t stream) {
    const float* x = (const float*)d_in[0];
    float*       y = (float*)d_out;
    (void)in_sizes; (void)n_in; (void)out_size; (void)d_ws; (void)ws_size;

    dim3 grid(NSEG * NTILE);   // 250 segments * 8 batch tiles = 2000 waves
    dim3 block(32);            // one wave32 per workgroup
    hipLaunchKernelGGL(bandpass_biquad_wmma, grid, block, 0, stream, x, y);
}